// ESN_26422638805330
// MI455X (gfx1250) — compile-verified
//
#include <hip/hip_runtime.h>
#include <hip/hip_bf16.h>
#include <math.h>

typedef float v2f __attribute__((ext_vector_type(2)));
typedef float v8f __attribute__((ext_vector_type(8)));

#define N_STEPS 8192
#define I_DIM   128
#define R_DIM   2048
#define O_DIM   10
#define KTOT    (I_DIM + R_DIM)     // 2176 concat K axis: [W_in | W] rows vs [u | x]
#define NWG     64                  // one WG per 32 reservoir rows
#define ROWS_PER_WG (R_DIM / NWG)   // 32
#define TPB     256                 // 8 wave32 per WG
#define NWAVES  (TPB / 32)
#define KPW     (KTOT / NWAVES)     // 272 K per wave (68 WMMA k-steps of 4)
#define PITCH   2180                // padded row pitch (floats): 8B aligned, bank-conflict free
#define LEAK    0.9f

// ---------------------------------------------------------------------------
// Persistent scan kernel: whole device cooperates on each sequential step.
// Each WG keeps its 32 rows of [W_in | W] in LDS (f32, ~279KB) for all 8192
// steps. Per step: stage [u_t | x_t] to LDS, WMMA matvec (K split over waves,
// two 16-row tiles sharing B loads, 4 independent WMMA accumulator chains),
// LDS reduce, tanh/leak update, fused partial readout via global f32 atomics,
// then a device-wide release/acquire barrier on a monotonic counter.
// ---------------------------------------------------------------------------
__global__ void __launch_bounds__(TPB, 1) esn_scan_kernel(
    const float* __restrict__ U, const float* __restrict__ Win,
    const float* __restrict__ W, const float* __restrict__ Wout,
    float* xbuf, float* out_acc, unsigned* counter)
{
    __shared__ float Wlds[ROWS_PER_WG * PITCH];   // 279,040 B
    __shared__ float xlds[PITCH];                 //   8,720 B
    __shared__ float ywave[NWAVES][ROWS_PER_WG];  //   1,024 B

    const int tid     = threadIdx.x;
    const int wg      = blockIdx.x;
    const int lane    = tid & 31;
    const int wave    = tid >> 5;
    const int rowbase = wg * ROWS_PER_WG;

    // ---- one-time: stage this WG's 32 rows of [W_in | W] into LDS ----
    for (int idx = tid; idx < ROWS_PER_WG * KTOT; idx += TPB) {
        int r = idx / KTOT, c = idx % KTOT;
        float v = (c < I_DIM) ? Win[(rowbase + r) * I_DIM + c]
                              : W[(rowbase + r) * R_DIM + (c - I_DIM)];
        Wlds[r * PITCH + c] = v;
    }
    // per-thread W_out column slice for the fused readout (threads 0..31)
    float wout[O_DIM];
    if (tid < ROWS_PER_WG) {
        #pragma unroll
        for (int o = 0; o < O_DIM; ++o) wout[o] = Wout[o * R_DIM + rowbase + tid];
    }
    __syncthreads();

    // per-lane WMMA addressing (ISA 16x16x4 f32 A layout):
    // lane L holds M = L%16; VGPR0/1 hold K = (0,1) for lanes<16, (2,3) for lanes>=16
    const int lm   = lane & 15;
    const int dsel = (lane >> 4) << 1;   // 0 or 2
    const int kw   = wave * KPW;
    const float* __restrict__ aptr0 = &Wlds[lm * PITCH + kw + dsel];        // tile rows 0-15
    const float* __restrict__ aptr1 = aptr0 + 16 * PITCH;                   // tile rows 16-31
    const float* __restrict__ bptr  = &xlds[kw + dsel];

    for (int s = 0; s < N_STEPS; ++s) {
        // make other WGs' xbuf stores visible (invalidate this wave's L0)
        __builtin_amdgcn_fence(__ATOMIC_ACQUIRE, "agent");

        // stage [u_s | x_s] into LDS
        const float* xsrc = xbuf + ((s - 1) & 1) * R_DIM;
        for (int i = tid; i < KTOT; i += TPB) {
            float v;
            if (i < I_DIM)   v = U[s * I_DIM + i];
            else if (s == 0) v = 0.0f;
            else             v = xsrc[i - I_DIM];
            xlds[i] = v;
        }
        __syncthreads();

        // y[rows] = [W_in|W] @ [u|x]; both 16-row tiles share each B load,
        // 4 independent accumulator chains -> no WMMA->WMMA RAW stalls.
        v8f acc0a = {0.f,0.f,0.f,0.f,0.f,0.f,0.f,0.f};
        v8f acc0b = acc0a, acc1a = acc0a, acc1b = acc0a;
        #pragma unroll 2
        for (int kk = 0; kk < KPW / 4; kk += 2) {
            const int o0 = 4 * kk, o1 = 4 * kk + 4;
            v2f b0  = *(const v2f*)(bptr  + o0);
            v2f b1  = *(const v2f*)(bptr  + o1);
            v2f a00 = *(const v2f*)(aptr0 + o0);
            v2f a01 = *(const v2f*)(aptr0 + o1);
            v2f a10 = *(const v2f*)(aptr1 + o0);
            v2f a11 = *(const v2f*)(aptr1 + o1);
            acc0a = __builtin_amdgcn_wmma_f32_16x16x4_f32(false, a00, false, b0,
                                                          (short)0, acc0a, false, false);
            acc1a = __builtin_amdgcn_wmma_f32_16x16x4_f32(false, a10, false, b0,
                                                          (short)0, acc1a, false, false);
            acc0b = __builtin_amdgcn_wmma_f32_16x16x4_f32(false, a01, false, b1,
                                                          (short)0, acc0b, false, false);
            acc1b = __builtin_amdgcn_wmma_f32_16x16x4_f32(false, a11, false, b1,
                                                          (short)0, acc1b, false, false);
        }
        v8f acc0 = acc0a + acc0b;   // tile 0: rows 0-15
        v8f acc1 = acc1a + acc1b;   // tile 1: rows 16-31

        // D layout: VGPR r -> M=r (lanes 0-15) / M=r+8 (lanes 16-31); cols identical
        if (lane == 0) {
            #pragma unroll
            for (int r = 0; r < 8; ++r) {
                ywave[wave][r]      = acc0[r];
                ywave[wave][16 + r] = acc1[r];
            }
        } else if (lane == 16) {
            #pragma unroll
            for (int r = 0; r < 8; ++r) {
                ywave[wave][8 + r]  = acc0[r];
                ywave[wave][24 + r] = acc1[r];
            }
        }
        __syncthreads();

        if (wave == 0) {   // 32 lanes: one reservoir row each
            float y = 0.f;
            #pragma unroll
            for (int w = 0; w < NWAVES; ++w) y += ywave[w][lane];
            const int   grow = rowbase + lane;
            const float xold = xlds[I_DIM + grow];
            const float a    = tanhf(y);
            const float xn   = (1.0f - LEAK) * xold + LEAK * a;
            xbuf[(s & 1) * R_DIM + grow] = xn;          // next step's state
            // fused readout: out_acc[s][o] += sum_rows Wout[o][row] * xn
            #pragma unroll
            for (int o = 0; o < O_DIM; ++o) {
                float p = wout[o] * xn;
                #pragma unroll
                for (int m = 16; m >= 1; m >>= 1) p += __shfl_xor(p, m, 32);
                if (lane == 0) atomicAdd(&out_acc[s * O_DIM + o], p);
            }
        }
        if (tid == 0) {
            // release our wave's stores, arrive, spin until all 64 WGs arrived
            __hip_atomic_fetch_add(counter, 1u, __ATOMIC_RELEASE,
                                   __HIP_MEMORY_SCOPE_AGENT);
            const unsigned target = (unsigned)(s + 1) * NWG;
            while (__hip_atomic_load(counter, __ATOMIC_ACQUIRE,
                                     __HIP_MEMORY_SCOPE_AGENT) < target)
                __builtin_amdgcn_s_sleep(1);
        }
        __syncthreads();
    }
}

__global__ void esn_init_kernel(unsigned* counter, float* xbuf, float* out_acc) {
    int i = blockIdx.x * blockDim.x + threadIdx.x;
    if (i == 0) *counter = 0u;
    if (i < 2 * R_DIM) xbuf[i] = 0.f;
    if (i < N_STEPS * O_DIM) out_acc[i] = 0.f;
}

__global__ void esn_out_kernel(const float* __restrict__ out_acc,
                               const float* __restrict__ b,
                               float* __restrict__ out) {
    int i = blockIdx.x * blockDim.x + threadIdx.x;
    if (i < N_STEPS * O_DIM) {
        float v = out_acc[i] + b[i % O_DIM];
        out[i] = 1.0f / (1.0f + expf(-v));
    }
}

extern "C" void kernel_launch(void* const* d_in, const int* in_sizes, int n_in,
                              void* d_out, int out_size, void* d_ws, size_t ws_size,
                              hipStream_t stream) {
    const float* U    = (const float*)d_in[0];
    const float* Win  = (const float*)d_in[1];
    const float* W    = (const float*)d_in[2];
    const float* Wout = (const float*)d_in[3];
    const float* b    = (const float*)d_in[4];
    float* out = (float*)d_out;

    char* ws = (char*)d_ws;
    unsigned* counter = (unsigned*)ws;                                   // 4 B
    float*    xbuf    = (float*)(ws + 256);                              // 2*2048 f32
    float*    out_acc = (float*)(ws + 256 + 2 * R_DIM * sizeof(float));  // 81920 f32

    const int n = N_STEPS * O_DIM;
    esn_init_kernel<<<(n + TPB - 1) / TPB, TPB, 0, stream>>>(counter, xbuf, out_acc);
    esn_scan_kernel<<<NWG, TPB, 0, stream>>>(U, Win, W, Wout, xbuf, out_acc, counter);
    esn_out_kernel<<<(n + TPB - 1) / TPB, TPB, 0, stream>>>(out_acc, b, out);
}